// MySmoothAP_50379966382606
// MI455X (gfx1250) — compile-verified
//
#include <hip/hip_runtime.h>

typedef __attribute__((ext_vector_type(2))) float v2f;
typedef __attribute__((ext_vector_type(8))) float v8f;

#define N_PTS     384
#define D_EMB     512
#define N_CLASSES 100

// ---------------------------------------------------------------------------
// Kernel 1: class id per row from one-hot labels [N, C]
// ---------------------------------------------------------------------------
__global__ void cls_kernel(const float* __restrict__ labels, int* __restrict__ cls) {
    int i = blockIdx.x * blockDim.x + threadIdx.x;
    if (i < N_PTS) {
        int c = 0;
        for (int j = 0; j < N_CLASSES; ++j)
            if (labels[i * N_CLASSES + j] > 0.5f) c = j;
        cls[i] = c;
    }
}

// ---------------------------------------------------------------------------
// Kernel 2: zero the accumulators (graph-capture safe init)
// ---------------------------------------------------------------------------
__global__ void init_kernel(float* __restrict__ acc) {
    acc[0] = 0.0f;  // sum of rk_pos/rk_all over target-true pairs
    acc[1] = 0.0f;  // count of target-true pairs
}

// ---------------------------------------------------------------------------
// Kernel 3: sim = E * E^T via V_WMMA_F32_16X16X4_F32.
// One wave32 per 16x16 output tile; K-loop over 512 in steps of 4.
//
// A fragment (16x4 f32): lanes 0-15 hold row M=lane, K={k0,k0+1};
//                        lanes 16-31 hold row M=lane-16, K={k0+2,k0+3}.
// B fragment (4x16 f32): same striping with N in place of M; since B = E^T,
//                        B[k][n] = E[n][k] -> identical addressing on row n.
// C/D (16x16 f32, 8 VGPRs): vgpr v, lane half h: D[M = v + 8h][N = lane&15].
// ---------------------------------------------------------------------------
__global__ void sim_wmma_kernel(const float* __restrict__ E, float* __restrict__ sim) {
    const int tm   = blockIdx.x;        // row tile (0..23)
    const int tn   = blockIdx.y;        // col tile (0..23)
    const int lane = threadIdx.x;       // 0..31 (wave32)
    const int half = lane >> 4;         // 0 or 1
    const int l16  = lane & 15;

    const float* rowA = E + (size_t)(tm * 16 + l16) * D_EMB;
    const float* rowB = E + (size_t)(tn * 16 + l16) * D_EMB;

    v8f c = {};
    #pragma unroll 4
    for (int k0 = 0; k0 < D_EMB; k0 += 4) {
        const int kk = k0 + 2 * half;
        v2f a = *(const v2f*)(rowA + kk);
        v2f b = *(const v2f*)(rowB + kk);
        // (neg_a, A, neg_b, B, c_mod, C, reuse_a, reuse_b)
        c = __builtin_amdgcn_wmma_f32_16x16x4_f32(false, a, false, b,
                                                  (short)0, c, false, false);
    }

    #pragma unroll
    for (int v = 0; v < 8; ++v) {
        sim[(size_t)(tm * 16 + v + 8 * half) * N_PTS + tn * 16 + l16] = c[v];
    }
}

// ---------------------------------------------------------------------------
// Kernel 4: the O(N^3) SmoothAP core. One block per query i, one thread per j.
// sim row i and the class-match mask are staged in LDS; the k-loop runs
// entirely out of LDS (VALU/transcendental bound, zero HBM traffic).
// Diagonal k==j handled exactly by subtracting sigmoid(0)=0.5 afterwards.
// ---------------------------------------------------------------------------
__global__ void smoothap_kernel(const float* __restrict__ sim,
                                const int* __restrict__ cls,
                                float* __restrict__ acc) {
    __shared__ float s[N_PTS];   // sim[i, :]
    __shared__ float t[N_PTS];   // target[i, :] as 0/1 float

    const int i  = blockIdx.x;
    const int ci = cls[i];

    for (int k = threadIdx.x; k < N_PTS; k += blockDim.x) {
        s[k] = sim[(size_t)i * N_PTS + k];
        t[k] = (cls[k] == ci) ? 1.0f : 0.0f;
    }
    __syncthreads();

    const int j  = threadIdx.x;          // blockDim.x == N_PTS
    const float sj = s[j];

    float rk_all = 0.0f;
    float rk_pos = 0.0f;
    #pragma unroll 4
    for (int k = 0; k < N_PTS; ++k) {
        float x = (s[k] - sj) * 100.0f;          // x / ANNEAL
        x = fminf(fmaxf(x, -50.0f), 50.0f);      // clip
        float e  = __expf(-x);                   // v_exp_f32 path
        float sg = __builtin_amdgcn_rcpf(1.0f + e);  // sigmoid
        rk_all += sg;
        rk_pos += sg * t[k];
    }
    // remove k == j term (mask = 1 - eye): sigmoid(0) == 0.5 exactly
    rk_all -= 0.5f;
    rk_pos -= 0.5f * t[j];

    if (t[j] > 0.5f) {
        float ratio = (rk_pos + 1.0f) / (rk_all + 1.0f);
        atomicAdd(&acc[0], ratio);
        atomicAdd(&acc[1], 1.0f);
    }
}

// ---------------------------------------------------------------------------
// Kernel 5: out = 1 - m_ap
// ---------------------------------------------------------------------------
__global__ void finalize_kernel(const float* __restrict__ acc, float* __restrict__ out) {
    out[0] = 1.0f - acc[0] / acc[1];
}

// ---------------------------------------------------------------------------
extern "C" void kernel_launch(void* const* d_in, const int* in_sizes, int n_in,
                              void* d_out, int out_size, void* d_ws, size_t ws_size,
                              hipStream_t stream) {
    const float* E      = (const float*)d_in[0];   // [384, 512] fp32
    const float* labels = (const float*)d_in[1];   // [384, 100] fp32 one-hot
    float*       out    = (float*)d_out;           // scalar fp32

    // workspace layout
    char* ws = (char*)d_ws;
    float* sim = (float*)ws;                                       // 384*384*4 B
    int*   cls = (int*)(ws + (size_t)N_PTS * N_PTS * sizeof(float));
    float* acc = (float*)(ws + (size_t)N_PTS * N_PTS * sizeof(float)
                             + (size_t)N_PTS * sizeof(int));       // 2 floats

    init_kernel<<<1, 1, 0, stream>>>(acc);
    cls_kernel<<<(N_PTS + 255) / 256, 256, 0, stream>>>(labels, cls);

    dim3 simGrid(N_PTS / 16, N_PTS / 16);   // 24 x 24 tiles, one wave32 each
    sim_wmma_kernel<<<simGrid, 32, 0, stream>>>(E, sim);

    smoothap_kernel<<<N_PTS, N_PTS, 0, stream>>>(sim, cls, acc);
    finalize_kernel<<<1, 1, 0, stream>>>(acc, out);
}